// PolyGraphConvolution_30743375904966
// MI455X (gfx1250) — compile-verified
//
#include <hip/hip_runtime.h>

typedef __attribute__((ext_vector_type(2))) float v2f;
typedef __attribute__((ext_vector_type(8))) float v8f;

#define IN_F   256
#define OUT_F  256
#define ADJ_POW 3

// ---------------------------------------------------------------------------
// Zero-fill the output (atomic accumulation target) — must run every call.
// ---------------------------------------------------------------------------
__global__ void __launch_bounds__(256) zero_kernel(float4* __restrict__ out, int n4) {
    int i = blockIdx.x * 256 + threadIdx.x;
    if (i < n4) out[i] = make_float4(0.f, 0.f, 0.f, 0.f);
}

// ---------------------------------------------------------------------------
// GEMM: h[16 rows x 256 cols per block] = x_tile @ W   using V_WMMA_F32_16X16X4_F32
//   block = 256 threads = 8 waves; wave w owns columns [w*32, w*32+32)
//   x tile staged in LDS with padded row stride (260) to avoid bank conflicts
// ---------------------------------------------------------------------------
__global__ void __launch_bounds__(256) gemm_kernel(const float* __restrict__ x,
                                                   const float* __restrict__ w,
                                                   float* __restrict__ h,
                                                   int n_nodes) {
    __shared__ float As[16 * 260];          // 16.6 KB, padded stride 260
    const int t  = threadIdx.x;
    const int m0 = blockIdx.x * 16;
    if (m0 + 16 > n_nodes) return;          // grid is exact (100000/16), uniform exit

    // Cooperative stage of x[m0..m0+16) x 256 into LDS (1024 float4 total)
    for (int i = t; i < 1024; i += 256) {
        const int r = i >> 6;               // row 0..15
        const int c = i & 63;               // float4 column 0..63
        const float4 v = ((const float4*)(x + (size_t)(m0 + r) * IN_F))[c];
        *((float4*)&As[r * 260 + c * 4]) = v;
    }
    __syncthreads();

    const int lane   = t & 31;
    const int laneLo = lane & 15;           // M (for A), N (for B/C/D)
    const int laneHi = lane >> 4;           // selects K-half / M-half
    const int n0     = (t >> 5) * 32;       // wave's column base

    v8f c0 = {};                            // cols [n0, n0+16)
    v8f c1 = {};                            // cols [n0+16, n0+32)

    const float* aP = &As[laneLo * 260];    // A row for this lane
    const float* wA = w + n0 + laneLo;      // B col for tile 0
    const float* wB = wA + 16;              // B col for tile 1

#pragma unroll 4
    for (int k0 = 0; k0 < IN_F; k0 += 4) {
        const int ka = k0 + laneHi * 2;     // lanes 16-31 take K-half {k0+2,k0+3}
        v2f a, b0, b1;
        // A 16x4 f32 layout: VGPR0 = K(ka), VGPR1 = K(ka+1)
        a.x = aP[ka];
        a.y = aP[ka + 1];
        // B 4x16 f32 layout: VGPR0 = row K(ka), VGPR1 = row K(ka+1), N = lane%16
        b0.x = wA[(size_t)ka * OUT_F];
        b0.y = wA[(size_t)(ka + 1) * OUT_F];
        b1.x = wB[(size_t)ka * OUT_F];
        b1.y = wB[(size_t)(ka + 1) * OUT_F];
        // 8-arg form: (neg_a, A, neg_b, B, c_mod, C, reuse_a, reuse_b)
        c0 = __builtin_amdgcn_wmma_f32_16x16x4_f32(false, a, false, b0,
                                                   (short)0, c0, false, false);
        c1 = __builtin_amdgcn_wmma_f32_16x16x4_f32(false, a, false, b1,
                                                   (short)0, c1, false, false);
    }

    // C/D layout: VGPR r -> row m0 + r (lanes 0-15) / m0 + r + 8 (lanes 16-31)
#pragma unroll
    for (int r = 0; r < 8; ++r) {
        const int m = m0 + r + laneHi * 8;
        h[(size_t)m * OUT_F + n0 + laneLo]      = c0[r];
        h[(size_t)m * OUT_F + n0 + 16 + laneLo] = c1[r];
    }
}

// ---------------------------------------------------------------------------
// Scatter: one wave per edge.  out[row] += val * h[col]   (256 f32 / edge)
// Each lane handles 8 consecutive floats: 2x b128 gather + 8 fire-and-forget
// global_atomic_add_f32 (relaxed, agent scope -> resolved at L2).
// ---------------------------------------------------------------------------
__global__ void __launch_bounds__(256) scatter_kernel(const float* __restrict__ h,
                                                      const int*   __restrict__ rows,
                                                      const int*   __restrict__ cols,
                                                      const float* __restrict__ vals,
                                                      float*       __restrict__ out,
                                                      int nnz) {
    const int e = blockIdx.x * 8 + (threadIdx.x >> 5);   // edge index (wave-uniform)
    if (e >= nnz) return;
    const int   lane = threadIdx.x & 31;
    const int   col  = cols[e];
    const int   row  = rows[e];
    const float val  = vals[e];

    const float4* hp = (const float4*)(h + (size_t)col * OUT_F) + lane * 2;
    const float4 a = hp[0];
    const float4 b = hp[1];

    float* o = out + (size_t)row * OUT_F + lane * 8;
    const float vbuf[8] = { a.x, a.y, a.z, a.w, b.x, b.y, b.z, b.w };
#pragma unroll
    for (int j = 0; j < 8; ++j) {
        __hip_atomic_fetch_add(o + j, vbuf[j] * val,
                               __ATOMIC_RELAXED, __HIP_MEMORY_SCOPE_AGENT);
    }
}

// ---------------------------------------------------------------------------
// Host entry: zero(out); for p in 0..2 { gemm(p) -> ws ; scatter(p) -> out }
// Same-stream launches give the GEMM->scatter dependency.  Workspace holds
// one h matrix: 100000*256*4 = 102.4 MB.
// ---------------------------------------------------------------------------
extern "C" void kernel_launch(void* const* d_in, const int* in_sizes, int n_in,
                              void* d_out, int out_size, void* d_ws, size_t ws_size,
                              hipStream_t stream) {
    const float* x      = (const float*)d_in[0];
    const float* weight = (const float*)d_in[1];
    const int*   erows  = (const int*)  d_in[2];
    const int*   ecols  = (const int*)  d_in[3];
    const float* evals  = (const float*)d_in[4];
    float*       out    = (float*)d_out;
    float*       h      = (float*)d_ws;

    const int n_nodes = in_sizes[0] / IN_F;       // 100000
    const int nnz     = in_sizes[2] / ADJ_POW;    // 3.2M

    const int n4 = out_size / 4;
    zero_kernel<<<(n4 + 255) / 256, 256, 0, stream>>>((float4*)out, n4);

    const int gemm_blocks = n_nodes / 16;         // 6250
    const int scat_blocks = (nnz + 7) / 8;        // 400000
    for (int p = 0; p < ADJ_POW; ++p) {
        gemm_kernel<<<gemm_blocks, 256, 0, stream>>>(
            x, weight + (size_t)p * IN_F * OUT_F, h, n_nodes);
        scatter_kernel<<<scat_blocks, 256, 0, stream>>>(
            h, erows + (size_t)p * nnz, ecols + (size_t)p * nnz,
            evals + (size_t)p * nnz, out, nnz);
    }
}